// ROM_52862457479542
// MI455X (gfx1250) — compile-verified
//
#include <hip/hip_runtime.h>
#include <math.h>

// NTM ROM.address fused kernel for MI455X (gfx1250, wave32).
// B=4096 batches, N=128 memory rows, M=64 key dim.
// Memory-bound (~1 FLOP/byte): one streaming pass of memory[] (134 MB) via
// coalesced global_load_b128 into padded LDS; per-row dot(k) and sum-of-
// squares on the WMMA pipe (V_WMMA_F32_16X16X4_F32); softmax / gate / shift /
// sharpen fused in the same block with wave32 shuffle reductions.

typedef __attribute__((ext_vector_type(2))) float v2f;
typedef __attribute__((ext_vector_type(8))) float v8f;

#define NROWS 128
#define MDIM  64
#define MPAD  68   // LDS row stride (floats); 68 % 64 == 4 -> conflict-free frag reads
#define NWAVES 8

__device__ __forceinline__ float waveReduceSum(float v) {
#pragma unroll
    for (int m = 16; m >= 1; m >>= 1) v += __shfl_xor(v, m, 32);
    return v;
}

__device__ __forceinline__ float waveReduceMax(float v) {
#pragma unroll
    for (int m = 16; m >= 1; m >>= 1) v = fmaxf(v, __shfl_xor(v, m, 32));
    return v;
}

// Two barriers per block-wide reduction: wave butterfly -> 8 partials in LDS ->
// every thread combines the 8 partials itself (no second tree).
__device__ __forceinline__ float blockReduceSum(float* red8, float v) {
    v = waveReduceSum(v);
    const int t = threadIdx.x;
    if ((t & 31) == 0) red8[t >> 5] = v;
    __syncthreads();
    float r = red8[0];
#pragma unroll
    for (int i = 1; i < NWAVES; ++i) r += red8[i];
    __syncthreads();   // red8 is reused by the next reduction
    return r;
}

__device__ __forceinline__ float blockReduceMax(float* red8, float v) {
    v = waveReduceMax(v);
    const int t = threadIdx.x;
    if ((t & 31) == 0) red8[t >> 5] = v;
    __syncthreads();
    float r = red8[0];
#pragma unroll
    for (int i = 1; i < NWAVES; ++i) r = fmaxf(r, red8[i]);
    __syncthreads();
    return r;
}

__device__ __forceinline__ float softplusf(float x) {
    return fmaxf(x, 0.0f) + log1pf(expf(-fabsf(x)));
}

__global__ __launch_bounds__(256) void rom_address_kernel(
    const float* __restrict__ memory,  // [B,N,M]
    const float* __restrict__ key,     // [B,M]
    const float* __restrict__ beta_,   // [B,1]
    const float* __restrict__ g_,      // [B,1]
    const float* __restrict__ s_,      // [B,3]
    const float* __restrict__ gamma_,  // [B,1]
    const float* __restrict__ w_prev,  // [B,N]
    float* __restrict__ out)           // [B,N]
{
    __shared__ float lmem[NROWS * MPAD];   // 34816 B
    __shared__ float lk[MDIM];
    __shared__ float ldot[NROWS];
    __shared__ float lsq[NROWS];
    __shared__ float lwg[NROWS];
    __shared__ float red8[NWAVES];

    const int b = blockIdx.x;
    const int t = threadIdx.x;

    // ---- Phase 1: coalesced stream of memory[b] (32 KB) into padded LDS ----
    const float4* gm = (const float4*)(memory + (size_t)b * NROWS * MDIM);
#pragma unroll
    for (int i = 0; i < 8; ++i) {
        int cid = t + i * 256;        // 0..2047 float4 chunks
        int row = cid >> 4;           // 16 float4 per row
        int off = cid & 15;
        float4 v = gm[cid];
        *(float4*)&lmem[row * MPAD + off * 4] = v;
    }
    if (t < MDIM) lk[t] = key[(size_t)b * MDIM + t];
    __syncthreads();

    // ---- Phase 2: WMMA dots + row sum-of-squares ----
    // Wave w owns rows [16w, 16w+16). A frag (16x4 f32): lane L holds
    // rows L%16, K = 2*(L/16) + {0,1}.  B frag: k-vector in column 0 only.
    const int lane = t & 31;
    const int wv   = t >> 5;          // 0..7
    const int r16  = lane & 15;
    const int half = lane >> 4;       // 0 or 1
    const float* arow = &lmem[(wv * 16 + r16) * MPAD + 2 * half];
    const float colSel = (r16 == 0) ? 1.0f : 0.0f;   // column 0 of B

    v8f acc_d = {0.f,0.f,0.f,0.f,0.f,0.f,0.f,0.f};
    v8f acc_q = {0.f,0.f,0.f,0.f,0.f,0.f,0.f,0.f};
#pragma unroll
    for (int m0 = 0; m0 < MDIM; m0 += 4) {
        v2f a;
        a.x = arow[m0]     + 1e-16f;   // mem + 1e-16 (reference)
        a.y = arow[m0 + 1] + 1e-16f;
        v2f a2; a2.x = a.x * a.x; a2.y = a.y * a.y;
        v2f bk;                        // kk = k + 1e-16 in column 0
        bk.x = colSel * (lk[m0 + 2 * half]     + 1e-16f);
        bk.y = colSel * (lk[m0 + 2 * half + 1] + 1e-16f);
        v2f bo; bo.x = colSel; bo.y = colSel;   // ones column for sumsq
        acc_d = __builtin_amdgcn_wmma_f32_16x16x4_f32(
                    false, a,  false, bk, (short)0, acc_d, false, false);
        acc_q = __builtin_amdgcn_wmma_f32_16x16x4_f32(
                    false, a2, false, bo, (short)0, acc_q, false, false);
    }
    // C/D f32 layout: VGPR r, lanes 0-15 -> D[M=r][N=lane]; lanes 16-31 -> D[M=r+8].
    // Column 0 lives in lanes 0 and 16.
    if (r16 == 0) {
        const int base = wv * 16 + half * 8;
#pragma unroll
        for (int r = 0; r < 8; ++r) {
            ldot[base + r] = acc_d[r];
            lsq [base + r] = acc_q[r];
        }
    }
    __syncthreads();

    // ---- Phase 3: fused addressing math (128 active rows) ----
    // key norm = ||k + 1e-16||
    float kv  = (t < MDIM) ? (lk[t] + 1e-16f) : 0.0f;
    float kn2 = blockReduceSum(red8, kv * kv);
    float key_norm = fmaxf(sqrtf(kn2), 1e-8f);

    // parameter squashing (redundant per thread; uniform -> scalarized loads)
    float beta  = softplusf(beta_[b]);
    float gg    = 1.0f / (1.0f + expf(-g_[b]));
    float s0r = s_[b * 3 + 0], s1r = s_[b * 3 + 1], s2r = s_[b * 3 + 2];
    float smax = fmaxf(s0r, fmaxf(s1r, s2r));
    float e0 = expf(s0r - smax), e1 = expf(s1r - smax), e2 = expf(s2r - smax);
    float einv = 1.0f / (e0 + e1 + e2);
    float s0 = e0 * einv, s1 = e1 * einv, s2 = e2 * einv;
    float gamma = 1.0f + softplusf(gamma_[b]);

    // content addressing: softmax(beta * cos)
    float a_val = -INFINITY;
    if (t < NROWS) {
        float rn = fmaxf(sqrtf(lsq[t]), 1e-8f);
        a_val = beta * (ldot[t] / (rn * key_norm));
    }
    float amax = blockReduceMax(red8, a_val);
    float ev = (t < NROWS) ? expf(a_val - amax) : 0.0f;
    float Z  = blockReduceSum(red8, ev);

    if (t < NROWS) {
        float wc = ev / Z;
        lwg[t] = gg * wc + (1.0f - gg) * w_prev[(size_t)b * NROWS + t];
    }
    __syncthreads();

    // circular 3-tap shift + sharpen
    float w = 0.0f;
    if (t < NROWS) {
        float wh = lwg[(t + NROWS - 1) & (NROWS - 1)] * s0
                 + lwg[t] * s1
                 + lwg[(t + 1) & (NROWS - 1)] * s2;
        w = powf(wh, gamma);
    }
    float S = blockReduceSum(red8, w);
    if (t < NROWS) out[(size_t)b * NROWS + t] = w / (S + 1e-16f);
}

extern "C" void kernel_launch(void* const* d_in, const int* in_sizes, int n_in,
                              void* d_out, int out_size, void* d_ws, size_t ws_size,
                              hipStream_t stream) {
    const float* memory = (const float*)d_in[0];
    const float* key    = (const float*)d_in[1];
    const float* beta   = (const float*)d_in[2];
    const float* g      = (const float*)d_in[3];
    const float* s      = (const float*)d_in[4];
    const float* gamma  = (const float*)d_in[5];
    const float* w_prev = (const float*)d_in[6];
    float* out = (float*)d_out;

    const int B = in_sizes[0] / (NROWS * MDIM);   // 4096
    rom_address_kernel<<<B, 256, 0, stream>>>(memory, key, beta, g, s, gamma,
                                              w_prev, out);
}